// VectorQuantizer_72928544686151
// MI455X (gfx1250) — compile-verified
//
#include <hip/hip_runtime.h>

typedef __attribute__((ext_vector_type(2))) float v2f;
typedef __attribute__((ext_vector_type(4))) float v4f;
typedef __attribute__((ext_vector_type(8))) float v8f;

#define E_DIM 32
#define WAVES 4
#define MT 4                 // 16-row tiles per block => 64 rows/block

// ---- pass 1: e_norm[i] = ||emb[i]||^2 (tiny; runs once per launch) ----
__global__ __launch_bounds__(256) void vq_enorm_kernel(
    const float* __restrict__ emb, float* __restrict__ enorm, int n_e) {
  const int i = blockIdx.x * blockDim.x + threadIdx.x;
  if (i < n_e) {
    const v4f* row = (const v4f*)(emb + (size_t)i * E_DIM);
    float s = 0.0f;
#pragma unroll
    for (int k = 0; k < 8; ++k) {
      const v4f v = row[k];
      s += v.x * v.x + v.y * v.y + v.z * v.z + v.w * v.w;
    }
    enorm[i] = s;
  }
}

// ---- pass 2: argmin over codebook via f32 WMMA ----
// dist(row, entry) = ||e||^2 - 2 <z_row, e>    (||z||^2 dropped: row-constant)
// 4 independent accumulator chains (one per row-tile) pipeline the XDL.
__global__ __launch_bounds__(32 * WAVES) void vq_argmin_kernel(
    const float* __restrict__ z,       // [rows, 32]
    const float* __restrict__ emb,     // [n_e, 32]
    const float* __restrict__ enorm,   // [n_e]
    float* __restrict__ zq_out,        // [rows, 32]
    float* __restrict__ idx_out,       // [rows] (written as float)
    int n_e, int index_offset)
{
  const int lane = threadIdx.x & 31;
  const int wave = threadIdx.x >> 5;
  const int row_base = blockIdx.x * (16 * MT);
  const int sub = lane & 15;   // row (A) / entry (B) / col (C) within tile
  const int hi  = lane >> 4;   // which K-pair this lane holds: {0,1} or {2,3}

  // A tiles: MT x (16x32) z block, ISA 16x4 f32 A layout, 8 K-steps
  v2f a[MT][8];
#pragma unroll
  for (int m = 0; m < MT; ++m) {
    const float* zrow = z + (size_t)(row_base + m * 16 + sub) * E_DIM + 2 * hi;
#pragma unroll
    for (int k = 0; k < 8; ++k)
      a[m][k] = *(const v2f*)(zrow + k * 4);
  }

  float run_val[MT][8];
  int   run_idx[MT][8];
#pragma unroll
  for (int m = 0; m < MT; ++m)
#pragma unroll
    for (int r = 0; r < 8; ++r) { run_val[m][r] = __builtin_inff(); run_idx[m][r] = 0; }

  const v8f vzero = {};
  const int col_tiles = n_e >> 4;

  for (int c = wave; c < col_tiles; c += WAVES) {
    const int entry = c * 16 + sub;
    const float* erow = emb + (size_t)entry * E_DIM + 2 * hi;

    v2f b[8];
#pragma unroll
    for (int k = 0; k < 8; ++k)
      b[k] = *(const v2f*)(erow + k * 4);
    const float en = enorm[entry];          // broadcast-friendly, L2-hot

    v8f acc[MT];
#pragma unroll
    for (int m = 0; m < MT; ++m) acc[m] = vzero;

    // k-outer / m-inner: dependent WMMAs are 4 apart -> XDL stays busy
#pragma unroll
    for (int k = 0; k < 8; ++k)
#pragma unroll
      for (int m = 0; m < MT; ++m)
        acc[m] = __builtin_amdgcn_wmma_f32_16x16x4_f32(
            /*neg_a=*/false, a[m][k], /*neg_b=*/false, b[k],
            /*c_mod=*/(short)0, acc[m], /*reuse_a=*/false, /*reuse_b=*/false);

#pragma unroll
    for (int m = 0; m < MT; ++m)
#pragma unroll
      for (int r = 0; r < 8; ++r) {
        const float dist = __builtin_fmaf(-2.0f, acc[m][r], en);
        if (dist < run_val[m][r]) { run_val[m][r] = dist; run_idx[m][r] = entry; }
      }
  }

  // butterfly argmin across the 16 columns (lanes within each 16-lane half)
#pragma unroll
  for (int msk = 1; msk <= 8; msk <<= 1) {
#pragma unroll
    for (int m = 0; m < MT; ++m)
#pragma unroll
      for (int r = 0; r < 8; ++r) {
        const float ov = __shfl_xor(run_val[m][r], msk, 32);
        const int   oi = __shfl_xor(run_idx[m][r], msk, 32);
        if (ov < run_val[m][r] || (ov == run_val[m][r] && oi < run_idx[m][r])) {
          run_val[m][r] = ov; run_idx[m][r] = oi;
        }
      }
  }

  __shared__ float s_val[WAVES][16 * MT];
  __shared__ int   s_idx[WAVES][16 * MT];
  if (sub == 0) {
#pragma unroll
    for (int m = 0; m < MT; ++m)
#pragma unroll
      for (int r = 0; r < 8; ++r) {
        const int row = m * 16 + r + 8 * hi;  // C layout: VGPR r, half -> row
        s_val[wave][row] = run_val[m][r];
        s_idx[wave][row] = run_idx[m][r];
      }
  }
  __syncthreads();

  // merge the 4 waves' candidates (disjoint column ranges), emit indices
  __shared__ int s_widx[16 * MT];
  if (threadIdx.x < 16 * MT) {
    float bv = s_val[0][threadIdx.x];
    int   bi = s_idx[0][threadIdx.x];
#pragma unroll
    for (int w = 1; w < WAVES; ++w) {
      const float v = s_val[w][threadIdx.x];
      const int   i = s_idx[w][threadIdx.x];
      if (v < bv || (v == bv && i < bi)) { bv = v; bi = i; }
    }
    s_widx[threadIdx.x] = bi;
    idx_out[row_base + threadIdx.x] = (float)(bi + index_offset);
  }
  __syncthreads();

  // gather z_q: 64 rows x 32 floats = 2048 floats; 128 threads x 16 floats
  {
    const int row = threadIdx.x >> 1;
    const int h2  = threadIdx.x & 1;
    const float* src = emb + (size_t)s_widx[row] * E_DIM + h2 * 16;
    float* dst = zq_out + (size_t)(row_base + row) * E_DIM + h2 * 16;
#pragma unroll
    for (int i = 0; i < 4; ++i)
      *(float4*)(dst + i * 4) = *(const float4*)(src + i * 4);
  }
}

extern "C" void kernel_launch(void* const* d_in, const int* in_sizes, int n_in,
                              void* d_out, int out_size, void* d_ws, size_t ws_size,
                              hipStream_t stream) {
  const float* z   = (const float*)d_in[0];
  const float* emb = (const float*)d_in[1];
  const int zn   = in_sizes[0];          // rows * 32
  const int rows = zn / E_DIM;           // 32768
  const int n_e  = in_sizes[1] / E_DIM;  // 8192

  float* zq_out  = (float*)d_out;        // output 0: z_q [rows*32]
  float* idx_out = (float*)d_out + zn;   // output 1: indices [rows]
  float* enorm   = (float*)d_ws;         // scratch: ||e||^2 [n_e]

  vq_enorm_kernel<<<(n_e + 255) / 256, 256, 0, stream>>>(emb, enorm, n_e);
  vq_argmin_kernel<<<rows / (16 * MT), 32 * WAVES, 0, stream>>>(
      z, emb, enorm, zq_out, idx_out, n_e, 32000);
}